// GraphNN_6803228197352
// MI455X (gfx1250) — compile-verified
//
#include <hip/hip_runtime.h>
#include <hip/hip_bf16.h>
#include <math.h>

#define NPTS 4096
#define CUT  3.6f

typedef __attribute__((ext_vector_type(2))) float v2f;
typedef __attribute__((ext_vector_type(8))) float v8f;

// D = A(16x4) * B(4x16) + C(16x16), all f32.
__device__ __forceinline__ v8f wmma4(v2f a, v2f b, v8f c) {
  return __builtin_amdgcn_wmma_f32_16x16x4_f32(
      /*neg_a=*/false, a, /*neg_b=*/false, b,
      /*c_mod=*/(short)0, c, /*reuse_a=*/false, /*reuse_b=*/false);
}

// ---------------------------------------------------------------------------
// Kernel 1: per-16-row tile, run the 3-layer MLP with f32 WMMA.
// Writes feat[N][16] and E[N] = exp(sum(keys*queries)).
// One wave (32 threads) per block -> EXEC all ones for WMMA.
// ---------------------------------------------------------------------------
__global__ __launch_bounds__(32) void mlp_kernel(
    const float* __restrict__ x,
    const float* __restrict__ W0, const float* __restrict__ b0,
    const float* __restrict__ W1, const float* __restrict__ b1,
    const float* __restrict__ W2, const float* __restrict__ b2,
    float* __restrict__ feat, float* __restrict__ E)
{
  __shared__ float xt[16][8];    // x tile, K padded 7->8
  __shared__ float w0t[8][16];   // W0^T (K-major), padded
  __shared__ float w1t[16][16];  // W1^T
  __shared__ float w2t[16][32];  // W2^T (two 16-wide halves)
  __shared__ float h[16][16];    // activation staging
  __shared__ float hb[16][16];   // keys|queries half

  const int lane = threadIdx.x;
  const int g = lane >> 4;       // lane group (0: lanes 0-15, 1: 16-31)
  const int n = lane & 15;       // column / M index within group
  const int row0 = blockIdx.x * 16;

  for (int idx = lane; idx < 16 * 8; idx += 32) {
    int m = idx >> 3, k = idx & 7;
    xt[m][k] = (k < 7) ? x[(row0 + m) * 7 + k] : 0.0f;
  }
  for (int idx = lane; idx < 8 * 16; idx += 32) {
    int k = idx >> 4, nn = idx & 15;
    w0t[k][nn] = (k < 7) ? W0[nn * 7 + k] : 0.0f;
  }
  for (int idx = lane; idx < 16 * 16; idx += 32) {
    int k = idx >> 4, nn = idx & 15;
    w1t[k][nn] = W1[nn * 16 + k];
  }
  for (int idx = lane; idx < 16 * 32; idx += 32) {
    int k = idx >> 5, nn = idx & 31;
    w2t[k][nn] = W2[nn * 16 + k];
  }
  __syncthreads();

  // Layer 0: h0 = atan(x @ W0^T + b0)   (K = 8 padded -> 2 WMMA steps)
  v8f c0 = {};
#pragma unroll
  for (int ks = 0; ks < 2; ++ks) {
    int kb = ks * 4;
    v2f a; a.x = xt[n][kb + 2 * g];     a.y = xt[n][kb + 2 * g + 1];
    v2f b; b.x = w0t[kb + 2 * g][n];    b.y = w0t[kb + 2 * g + 1][n];
    c0 = wmma4(a, b, c0);
  }
#pragma unroll
  for (int v = 0; v < 8; ++v) h[v + 8 * g][n] = atanf(c0[v] + b0[n]);
  __syncthreads();

  // Layer 1: h1 = atan(h0 @ W1^T + b1)   (K = 16 -> 4 WMMA steps)
  v8f c1 = {};
#pragma unroll
  for (int ks = 0; ks < 4; ++ks) {
    int kb = ks * 4;
    v2f a; a.x = h[n][kb + 2 * g];      a.y = h[n][kb + 2 * g + 1];
    v2f b; b.x = w1t[kb + 2 * g][n];    b.y = w1t[kb + 2 * g + 1][n];
    c1 = wmma4(a, b, c1);
  }
  __syncthreads();
#pragma unroll
  for (int v = 0; v < 8; ++v) h[v + 8 * g][n] = atanf(c1[v] + b1[n]);
  __syncthreads();

  // Layer 2: h2 = h1 @ W2^T + b2  (32 outputs -> two WMMA chains)
  v8f c2a = {}, c2b = {};
#pragma unroll
  for (int ks = 0; ks < 4; ++ks) {
    int kb = ks * 4;
    v2f a;  a.x  = h[n][kb + 2 * g];         a.y  = h[n][kb + 2 * g + 1];
    v2f ba; ba.x = w2t[kb + 2 * g][n];       ba.y = w2t[kb + 2 * g + 1][n];
    v2f bb; bb.x = w2t[kb + 2 * g][n + 16];  bb.y = w2t[kb + 2 * g + 1][n + 16];
    c2a = wmma4(a, ba, c2a);
    c2b = wmma4(a, bb, c2b);
  }
#pragma unroll
  for (int v = 0; v < 8; ++v) {
    feat[(size_t)(row0 + v + 8 * g) * 16 + n] = c2a[v] + b2[n];   // feat half
    hb[v + 8 * g][n] = c2b[v] + b2[16 + n];                       // keys|queries
  }
  __syncthreads();

  if (lane < 16) {
    float s = 0.0f;
#pragma unroll
    for (int d = 0; d < 8; ++d) s += hb[lane][d] * hb[lane][d + 8];
    E[row0 + lane] = expf(s);   // softmax shift cancels exactly in this problem
  }
}

// ---------------------------------------------------------------------------
// Kernel 2: firstk[j] = min k with L1(coords[k], coords[j]) <= CUT.
// (filled[i][j] == (i >= firstk[j]); firstk[j] <= j always since dist(j,j)=0)
// ---------------------------------------------------------------------------
__global__ __launch_bounds__(256) void firstk_kernel(
    const float* __restrict__ x, int* __restrict__ firstk)
{
  __shared__ float cx[256], cy[256], cz[256];
  const int j = blockIdx.x * 256 + threadIdx.x;
  const float px = x[j * 7 + 0], py = x[j * 7 + 1], pz = x[j * 7 + 2];
  int found = -1;
  for (int kb = 0; kb < NPTS; kb += 256) {
    int kk = kb + threadIdx.x;
    cx[threadIdx.x] = x[kk * 7 + 0];
    cy[threadIdx.x] = x[kk * 7 + 1];
    cz[threadIdx.x] = x[kk * 7 + 2];
    __syncthreads();
    if (found < 0) {
      for (int t = 0; t < 256; ++t) {
        float d = fabsf(px - cx[t]) + fabsf(py - cy[t]) + fabsf(pz - cz[t]);
        if (d <= CUT) { found = kb + t; break; }
      }
    }
    if (__syncthreads_count(found >= 0) == 256) break;  // uniform exit
  }
  firstk[j] = (found < 0) ? j : found;
}

// ---------------------------------------------------------------------------
// Kernel 3: zero bucket arrays (ws is poisoned, must re-zero every call)
// ---------------------------------------------------------------------------
__global__ void zero_kernel(float* __restrict__ buf, int n) {
  int i = blockIdx.x * blockDim.x + threadIdx.x;
  if (i < n) buf[i] = 0.0f;
}

// ---------------------------------------------------------------------------
// Kernel 4: bucket accumulate by t = firstk[j]:
//   p[t][d] += E[j]*feat[j][d],  q[t] += E[j],  c[t] += 1
// ---------------------------------------------------------------------------
__global__ __launch_bounds__(256) void bucket_kernel(
    const float* __restrict__ feat, const float* __restrict__ E,
    const int* __restrict__ firstk,
    float* __restrict__ p, float* __restrict__ q, float* __restrict__ c)
{
  int j = blockIdx.x * 256 + threadIdx.x;
  int t = firstk[j];
  float e = E[j];
  atomicAdd(&q[t], e);
  atomicAdd(&c[t], 1.0f);
#pragma unroll
  for (int d = 0; d < 16; ++d)
    atomicAdd(&p[(size_t)t * 16 + d], e * feat[(size_t)j * 16 + d]);
}

// ---------------------------------------------------------------------------
// Kernel 5: inclusive prefix sum over t=0..4095 for 18 channels
// (one block per channel; 1024 threads x 4 elems; Hillis-Steele in LDS)
// ---------------------------------------------------------------------------
__global__ __launch_bounds__(1024) void scan_kernel(
    float* __restrict__ p, float* __restrict__ q, float* __restrict__ c)
{
  __shared__ float part[1024];
  const int ch = blockIdx.x;
  float* base; int stride;
  if (ch < 16)      { base = p + ch; stride = 16; }
  else if (ch == 16){ base = q;      stride = 1;  }
  else              { base = c;      stride = 1;  }

  const int t0 = threadIdx.x * 4;
  float a0 = base[(size_t)(t0 + 0) * stride];
  float a1 = a0 + base[(size_t)(t0 + 1) * stride];
  float a2 = a1 + base[(size_t)(t0 + 2) * stride];
  float a3 = a2 + base[(size_t)(t0 + 3) * stride];
  part[threadIdx.x] = a3;
  __syncthreads();
  for (int off = 1; off < 1024; off <<= 1) {
    float t = (threadIdx.x >= off) ? part[threadIdx.x - off] : 0.0f;
    __syncthreads();
    part[threadIdx.x] += t;
    __syncthreads();
  }
  float excl = (threadIdx.x > 0) ? part[threadIdx.x - 1] : 0.0f;
  base[(size_t)(t0 + 0) * stride] = a0 + excl;
  base[(size_t)(t0 + 1) * stride] = a1 + excl;
  base[(size_t)(t0 + 2) * stride] = a2 + excl;
  base[(size_t)(t0 + 3) * stride] = a3 + excl;
}

// ---------------------------------------------------------------------------
// Kernel 6: my_input[i] = P_i / (Q_i + (N - C_i));
//           out = atan(my_input @ We^T + be) @ Wd^T + bd   (WMMA again)
// ---------------------------------------------------------------------------
__global__ __launch_bounds__(32) void out_kernel(
    const float* __restrict__ p, const float* __restrict__ q,
    const float* __restrict__ c,
    const float* __restrict__ We, const float* __restrict__ be,
    const float* __restrict__ Wd, const float* __restrict__ bd,
    float* __restrict__ out)
{
  __shared__ float mi[16][16];
  __shared__ float wet[16][16];
  __shared__ float wdt[16][16];  // Wd^T padded 7->16 cols
  __shared__ float cod[16][16];

  const int lane = threadIdx.x;
  const int g = lane >> 4;
  const int n = lane & 15;
  const int row0 = blockIdx.x * 16;

  for (int idx = lane; idx < 256; idx += 32) {
    int m = idx >> 4, d = idx & 15;
    int i = row0 + m;
    float denom = q[i] + ((float)NPTS - c[i]);
    mi[m][d] = p[(size_t)i * 16 + d] / denom;
  }
  for (int idx = lane; idx < 256; idx += 32) {
    int k = idx >> 4, nn = idx & 15;
    wet[k][nn] = We[nn * 16 + k];
    wdt[k][nn] = (nn < 7) ? Wd[nn * 16 + k] : 0.0f;
  }
  __syncthreads();

  v8f cc = {};
#pragma unroll
  for (int ks = 0; ks < 4; ++ks) {
    int kb = ks * 4;
    v2f a; a.x = mi[n][kb + 2 * g];   a.y = mi[n][kb + 2 * g + 1];
    v2f b; b.x = wet[kb + 2 * g][n];  b.y = wet[kb + 2 * g + 1][n];
    cc = wmma4(a, b, cc);
  }
#pragma unroll
  for (int v = 0; v < 8; ++v) cod[v + 8 * g][n] = atanf(cc[v] + be[n]);
  __syncthreads();

  v8f co = {};
#pragma unroll
  for (int ks = 0; ks < 4; ++ks) {
    int kb = ks * 4;
    v2f a; a.x = cod[n][kb + 2 * g];  a.y = cod[n][kb + 2 * g + 1];
    v2f b; b.x = wdt[kb + 2 * g][n];  b.y = wdt[kb + 2 * g + 1][n];
    co = wmma4(a, b, co);
  }
  if (n < 7) {
#pragma unroll
    for (int v = 0; v < 8; ++v)
      out[(size_t)(row0 + v + 8 * g) * 7 + n] = co[v] + bd[n];
  }
}

// ---------------------------------------------------------------------------
extern "C" void kernel_launch(void* const* d_in, const int* in_sizes, int n_in,
                              void* d_out, int out_size, void* d_ws, size_t ws_size,
                              hipStream_t stream)
{
  const float* x  = (const float*)d_in[0];
  const float* W0 = (const float*)d_in[1];
  const float* b0 = (const float*)d_in[2];
  const float* W1 = (const float*)d_in[3];
  const float* b1 = (const float*)d_in[4];
  const float* W2 = (const float*)d_in[5];
  const float* b2 = (const float*)d_in[6];
  const float* We = (const float*)d_in[7];
  const float* be = (const float*)d_in[8];
  const float* Wd = (const float*)d_in[9];
  const float* bd = (const float*)d_in[10];
  float* out = (float*)d_out;

  // Workspace layout (floats): feat[N*16] | E[N] | firstk[N](int) | p[N*16] | q[N] | c[N]
  float* feat = (float*)d_ws;
  float* E    = feat + (size_t)NPTS * 16;
  int*   fk   = (int*)(E + NPTS);
  float* p    = (float*)(fk + NPTS);
  float* q    = p + (size_t)NPTS * 16;
  float* c    = q + NPTS;
  (void)ws_size; (void)in_sizes; (void)n_in; (void)out_size;

  mlp_kernel<<<NPTS / 16, 32, 0, stream>>>(x, W0, b0, W1, b1, W2, b2, feat, E);
  firstk_kernel<<<NPTS / 256, 256, 0, stream>>>(x, fk);
  zero_kernel<<<(NPTS * 18 + 255) / 256, 256, 0, stream>>>(p, NPTS * 18); // p,q,c contiguous
  bucket_kernel<<<NPTS / 256, 256, 0, stream>>>(feat, E, fk, p, q, c);
  scan_kernel<<<18, 1024, 0, stream>>>(p, q, c);
  out_kernel<<<NPTS / 16, 32, 0, stream>>>(p, q, c, We, be, Wd, bd, out);
}